// InteractiveAttention_26276609916991
// MI455X (gfx1250) — compile-verified
//
#include <hip/hip_runtime.h>
#include <cmath>

// Problem constants (from reference setup_inputs)
#define BATCH 64
#define LC    512
#define LA    32
#define DIM   768
#define FEPS  1e-7f

typedef __attribute__((ext_vector_type(2))) float v2f;
typedef __attribute__((ext_vector_type(8))) float v8f;

// -------------------------------------------------------------------------
// Kernel A: asp_avg[b,d] = mean_la asp_text[b,la,d]     ([64,32,768]->[64,768])
// grid = (BATCH*DIM)/256 = 192 blocks of 256; consecutive threads -> consecutive d
// -------------------------------------------------------------------------
__global__ __launch_bounds__(256) void ia_asp_avg_kernel(
    const float* __restrict__ asp_text,   // [B, LA, D]
    float* __restrict__ asp_avg)          // [B, D]
{
    const int i = blockIdx.x * 256 + threadIdx.x;   // i = b*DIM + d
    const int b = i / DIM;
    const int d = i - b * DIM;
    const float* p = asp_text + (size_t)b * LA * DIM + d;
    float s = 0.0f;
#pragma unroll 8
    for (int la = 0; la < LA; ++la)
        s += p[(size_t)la * DIM];
    asp_avg[i] = s * (1.0f / (float)LA);
}

// -------------------------------------------------------------------------
// Kernel B: v = asp_avg @ W^T   (M=64, N=768, K=768), fp32 WMMA 16x16x4
//   v[b,d] = sum_e W[d,e] * asp_avg[b,e]  => B-matrix[k,n] = W[n,k]
// One wave (32 threads) per 16x16 output tile; 192 WMMA ops per tile.
// fp32 A 16x4 layout: lane&15 = M row, lane>>4 selects K pair {0,1}/{2,3};
// fp32 B 4x16 layout mirrors A (lane&15 = N col); C/D: vgpr i -> M = i + 8*(lane>>4).
// -------------------------------------------------------------------------
__global__ __launch_bounds__(32) void ia_proj_wmma_kernel(
    const float* __restrict__ asp_avg,   // [B, D]  (A matrix, row-major)
    const float* __restrict__ W,         // [D, D]  row-major W[d,e]
    float* __restrict__ vmat)            // [B, D]
{
    const int nTile = blockIdx.x;        // 0..47  (D/16)
    const int mTile = blockIdx.y;        // 0..3   (B/16)
    const int lane  = threadIdx.x;       // 0..31
    const int r     = lane & 15;         // row (A) / col (B,C,D)
    const int h     = lane >> 4;         // K-half selector

    const int m0 = mTile * 16;
    const int n0 = nTile * 16;

    const float* aRow = asp_avg + (size_t)(m0 + r) * DIM;  // A row for this lane
    const float* bRow = W       + (size_t)(n0 + r) * DIM;  // W row n -> B column n

    v8f c = {};
    for (int k = 0; k < DIM; k += 4) {
        const int kk = k + 2 * h;
        v2f a, bf;
        a.x  = aRow[kk];
        a.y  = aRow[kk + 1];
        bf.x = bRow[kk];          // B[kk,   n0+r] = W[n0+r, kk]
        bf.y = bRow[kk + 1];      // B[kk+1, n0+r]
        // 8 args: (neg_a, A, neg_b, B, c_mod, C, reuse_a, reuse_b)
        c = __builtin_amdgcn_wmma_f32_16x16x4_f32(
                false, a, false, bf, (short)0, c, false, false);
    }

#pragma unroll
    for (int i = 0; i < 8; ++i) {
        const int m = h * 8 + i;
        vmat[(size_t)(m0 + m) * DIM + n0 + r] = c[i];
    }
}

// -------------------------------------------------------------------------
// Kernel C: per-batch fused  scores -> tanh/exp/mask -> normalize -> pool.
// One 256-thread (8-wave) workgroup per batch. Pass 1 over context is the
// HBM-bound read; pass 2 (pooling) re-reads the same 1.5 MB slice from L2.
// -------------------------------------------------------------------------
__global__ __launch_bounds__(256) void ia_attend_kernel(
    const float* __restrict__ context,   // [B, LC, D]
    const float* __restrict__ vmat,      // [B, D]
    const float* __restrict__ bias,      // [LC]
    const int*   __restrict__ mask,      // [B, LC] (bool widened to int)
    float* __restrict__ out)             // [B, D]
{
    const int b    = blockIdx.x;
    const int tid  = threadIdx.x;        // 0..255
    const int lane = tid & 31;
    const int wave = tid >> 5;           // 0..7

    __shared__ float v_s[DIM];
    __shared__ float a_s[LC];
    __shared__ float red_s[8];

    const float* ctx_b = context + (size_t)b * LC * DIM;

    // stage v[b,:] in LDS
    for (int d = tid; d < DIM; d += 256)
        v_s[d] = vmat[(size_t)b * DIM + d];
    __syncthreads();

    // scores: one wave per context row, coalesced 128B reads, wave32 reduce
    for (int l = wave; l < LC; l += 8) {
        const float* row = ctx_b + (size_t)l * DIM;
        if (l + 8 < LC)
            __builtin_prefetch(row + 8 * DIM, 0, 0);   // global_prefetch_b8
        float partial = 0.0f;
#pragma unroll 4
        for (int d = lane; d < DIM; d += 32)
            partial += row[d] * v_s[d];
#pragma unroll
        for (int off = 16; off > 0; off >>= 1)
            partial += __shfl_down(partial, off, 32);
        if (lane == 0) {
            float s = tanhf(partial + bias[l]);
            float a = expf(s);
            a *= (mask[(size_t)b * LC + l] != 0) ? 1.0f : 0.0f;
            a_s[l] = a;
        }
    }
    __syncthreads();

    // block-wide sum of a_s[0..LC)
    float psum = 0.0f;
    for (int l = tid; l < LC; l += 256)
        psum += a_s[l];
#pragma unroll
    for (int off = 16; off > 0; off >>= 1)
        psum += __shfl_down(psum, off, 32);
    if (lane == 0) red_s[wave] = psum;
    __syncthreads();

    float total = 0.0f;
#pragma unroll
    for (int w = 0; w < 8; ++w) total += red_s[w];
    const float inv = 1.0f / (total + FEPS);

    // pooling: thread owns d = tid, tid+256, tid+512; rows stream from L2
    float acc0 = 0.0f, acc1 = 0.0f, acc2 = 0.0f;
    for (int l = 0; l < LC; ++l) {
        const float a = a_s[l];
        const float* row = ctx_b + (size_t)l * DIM;
        acc0 += a * row[tid];
        acc1 += a * row[tid + 256];
        acc2 += a * row[tid + 512];
    }
    out[(size_t)b * DIM + tid]       = acc0 * inv;
    out[(size_t)b * DIM + tid + 256] = acc1 * inv;
    out[(size_t)b * DIM + tid + 512] = acc2 * inv;
}

// -------------------------------------------------------------------------
extern "C" void kernel_launch(void* const* d_in, const int* in_sizes, int n_in,
                              void* d_out, int out_size, void* d_ws, size_t ws_size,
                              hipStream_t stream) {
    (void)in_sizes; (void)n_in; (void)out_size; (void)ws_size;

    const float* context   = (const float*)d_in[0];   // [64,512,768]
    const float* asp_text  = (const float*)d_in[1];   // [64,32,768]
    const float* context_w = (const float*)d_in[2];   // [768,768]
    const float* context_b = (const float*)d_in[3];   // [512]
    const int*   ctx_mask  = (const int*)  d_in[4];   // [64,512]
    float*       out       = (float*)d_out;           // [64,768]

    char* ws = (char*)d_ws;
    float* asp_avg = (float*)(ws);                            // 64*768*4 = 192 KB
    float* vmat    = (float*)(ws + (size_t)BATCH * DIM * 4);  // 64*768*4 = 192 KB

    // A: aspect average
    ia_asp_avg_kernel<<<dim3((BATCH * DIM) / 256), dim3(256), 0, stream>>>(
        asp_text, asp_avg);

    // B: v = asp_avg @ W^T via fp32 WMMA 16x16x4 (one wave per 16x16 tile)
    ia_proj_wmma_kernel<<<dim3(DIM / 16, BATCH / 16), dim3(32), 0, stream>>>(
        asp_avg, context_w, vmat);

    // C: fused scores + masked-exp-tanh softmax + weighted pooling
    ia_attend_kernel<<<dim3(BATCH), dim3(256), 0, stream>>>(
        context, vmat, context_b, ctx_mask, out);
}